// SimpleGCN_45707041964395
// MI455X (gfx1250) — compile-verified
//
#include <hip/hip_runtime.h>
#include <hip/hip_bf16.h>

// ---------------------------------------------------------------------------
// SimpleGCN on MI455X (gfx1250, wave32).
// Memory/atomic-bound (edge scatter ~3.3GB; working set ~154MB fits in 192MB
// L2). GEMMs (6.6 GFLOP) use V_WMMA_F32_16X16X4_F32 (fp32-exact, fully
// unrolled 32x K-loop). A-tiles staged with GLOBAL_LOAD_ASYNC_TO_LDS_B128
// (ASYNCcnt DMA) + LDS-resident aggregate combine.
// ---------------------------------------------------------------------------

#define N_NODES 100000
#define N_EDGES 1600000
#define D       128
#define ND      ((size_t)N_NODES * D)

typedef __attribute__((ext_vector_type(2))) float v2f;
typedef __attribute__((ext_vector_type(8))) float v8f;

// ---------------- zero a region (float4 granularity) -----------------------
__global__ void k_zero(float4* __restrict__ p, long n4) {
    long i = (long)blockIdx.x * blockDim.x + threadIdx.x;
    long stride = (long)gridDim.x * blockDim.x;
    float4 z; z.x = z.y = z.z = z.w = 0.0f;
    for (; i < n4; i += stride) p[i] = z;
}

// ---------------- degree: atomicAdd 1.0 per edge on dst --------------------
__global__ void k_degree(const int* __restrict__ ei, float* __restrict__ deg) {
    long i = (long)blockIdx.x * blockDim.x + threadIdx.x;
    long stride = (long)gridDim.x * blockDim.x;
    for (; i < N_EDGES; i += stride) {
        int d = ei[N_EDGES + i];
        atomicAdd(&deg[d], 1.0f);
    }
}

// ---------------- edge scatter: one wave32 per edge ------------------------
// lane loads one float4 (128 floats / 32 lanes), 4 float atomics to dst row.
__global__ __launch_bounds__(256)
void k_scatter(const float* __restrict__ x, const int* __restrict__ ei,
               float* __restrict__ msg) {
    long wid = ((long)blockIdx.x * blockDim.x + threadIdx.x) >> 5;
    int lane = threadIdx.x & 31;
    if (wid >= N_EDGES) return;
    int s = ei[wid];
    int d = ei[N_EDGES + wid];
    const float4* srow = (const float4*)(x + (size_t)s * D);
    float4 v = srow[lane];
    float* drow = msg + (size_t)d * D + lane * 4;
    atomicAdd(drow + 0, v.x);
    atomicAdd(drow + 1, v.y);
    atomicAdd(drow + 2, v.z);
    atomicAdd(drow + 3, v.w);
}

// ---- async DMA of one 16x128 fp32 tile (8KB) from global into LDS ---------
// 512 x 16B per-lane transfers, 2 per thread (256 threads). ASYNCcnt-tracked.
__device__ __forceinline__ void async_tile_load(const float* __restrict__ g,
                                                float* s) {
    int tid = threadIdx.x;
    #pragma unroll
    for (int j = 0; j < 2; ++j) {
        int i = tid + j * 256;
        unsigned lds = (unsigned)(unsigned long long)(s + i * 4);
        unsigned long long ga = (unsigned long long)(g + i * 4);
        asm volatile("global_load_async_to_lds_b128 %0, %1, off"
                     :: "v"(lds), "v"(ga) : "memory");
    }
}

// ---- stage x-tile into sNF: sNF = nf_tile + msg_tile * rdeg(row) ----------
__device__ __forceinline__ void stage_tile(const float* __restrict__ nf,
                                           const float* __restrict__ msg,
                                           const float* __restrict__ deg,
                                           int m0, float* sNF, float* sMSG,
                                           float* sR) {
    const int tid = threadIdx.x;
    if (tid < 16) {
        float dg = deg[m0 + tid];
        sR[tid] = 1.0f / (dg < 1.0f ? 1.0f : dg);
    }
    const size_t goff = (size_t)m0 * D;
    async_tile_load(nf + goff, sNF);
    async_tile_load(msg + goff, sMSG);
    asm volatile("s_wait_asynccnt 0x0" ::: "memory");
    __syncthreads();
    // combine in LDS with b128 traffic: 512 float4s, 2 per thread
    #pragma unroll
    for (int j = 0; j < 2; ++j) {
        int i = tid + j * 256;
        int r = i >> 5;                     // (i*4)/128
        float rd = sR[r];
        float4 a = ((float4*)sNF)[i];
        float4 m = ((float4*)sMSG)[i];
        a.x += m.x * rd; a.y += m.y * rd; a.z += m.z * rd; a.w += m.w * rd;
        ((float4*)sNF)[i] = a;
    }
    __syncthreads();
}

// ---- 16x16 tile GEMM body: c += A(16x128) * W(:,col-slice) via 32 WMMA ----
__device__ __forceinline__ v8f wmma_tile(const float* sA,
                                         const float* __restrict__ W,
                                         int l16, int half, int col) {
    v8f c = {};
    #pragma unroll
    for (int k = 0; k < D; k += 4) {
        // A 16x4 fp32: V0 -> K=k+2h, V1 -> K=k+2h+1, M=l16 (ISA 7.12.2)
        v2f a, b;
        const int ka = k + 2 * half;
        a.x = sA[l16 * D + ka];
        a.y = sA[l16 * D + ka + 1];
        // B 4x16 fp32: V_j, half h -> K = k + j + 2h, N = col
        b.x = W[(size_t)(k + 2 * half) * D + col];
        b.y = W[(size_t)(k + 2 * half + 1) * D + col];
        c = __builtin_amdgcn_wmma_f32_16x16x4_f32(
                false, a, false, b, (short)0, c, false, false);
    }
    return c;
}

// ---------------- fused aggregate + linear + relu (layer 1) ----------------
// Block = 256 thr = 8 waves; block computes a 16x128 output tile; each wave
// owns a 16-column slice of W.
__global__ __launch_bounds__(256)
void k_layer1(const float* __restrict__ nf, const float* __restrict__ msg,
              const float* __restrict__ deg, const float* __restrict__ W,
              const float* __restrict__ bias, float* __restrict__ out) {
    __shared__ float sNF[16 * D];
    __shared__ float sMSG[16 * D];
    __shared__ float sR[16];
    const int m0 = blockIdx.x * 16;
    stage_tile(nf, msg, deg, m0, sNF, sMSG, sR);

    const int wave = threadIdx.x >> 5;
    const int lane = threadIdx.x & 31;
    const int half = lane >> 4;
    const int l16  = lane & 15;
    const int col  = wave * 16 + l16;

    v8f c = wmma_tile(sNF, W, l16, half, col);

    const float bv = bias[col];
    #pragma unroll
    for (int v = 0; v < 8; ++v) {
        int row = v + half * 8;   // C/D: VGPR v -> M=v (lanes 16-31: +8)
        float val = c[v] + bv;
        out[(size_t)(m0 + row) * D + col] = val > 0.0f ? val : 0.0f;
    }
}

// ------- fused aggregate + linear + relu + mean-readout (layer 2) ----------
__global__ __launch_bounds__(256)
void k_layer2(const float* __restrict__ h1, const float* __restrict__ msg,
              const float* __restrict__ deg, const float* __restrict__ W,
              const float* __restrict__ bias, float* __restrict__ colsum) {
    __shared__ float sNF[16 * D];
    __shared__ float sMSG[16 * D];
    __shared__ float sR[16];
    const int m0 = blockIdx.x * 16;
    stage_tile(h1, msg, deg, m0, sNF, sMSG, sR);

    const int wave = threadIdx.x >> 5;
    const int lane = threadIdx.x & 31;
    const int half = lane >> 4;
    const int l16  = lane & 15;
    const int col  = wave * 16 + l16;

    v8f c = wmma_tile(sNF, W, l16, half, col);

    // relu + partial column sum over this lane's 8 rows -> one atomic
    const float bv = bias[col];
    float s = 0.0f;
    #pragma unroll
    for (int v = 0; v < 8; ++v) {
        float val = c[v] + bv;
        s += val > 0.0f ? val : 0.0f;
    }
    atomicAdd(&colsum[col], s);
}

// ---------------- readout: out = (colsum/N) @ W3 + b3 ----------------------
__global__ void k_final(const float* __restrict__ colsum,
                        const float* __restrict__ W3,
                        const float* __restrict__ b3,
                        float* __restrict__ out) {
    int j = threadIdx.x;              // 128 threads
    const float inv = 1.0f / (float)N_NODES;
    float acc = b3[j];
    #pragma unroll 8
    for (int k = 0; k < D; ++k)
        acc += (colsum[k] * inv) * W3[(size_t)k * D + j];
    out[j] = acc;
}

// ---------------------------------------------------------------------------
extern "C" void kernel_launch(void* const* d_in, const int* in_sizes, int n_in,
                              void* d_out, int out_size, void* d_ws, size_t ws_size,
                              hipStream_t stream) {
    const float* nf = (const float*)d_in[0];
    const int*   ei = (const int*)d_in[1];
    const float* W1 = (const float*)d_in[2];
    const float* b1 = (const float*)d_in[3];
    const float* W2 = (const float*)d_in[4];
    const float* b2 = (const float*)d_in[5];
    const float* W3 = (const float*)d_in[6];
    const float* b3 = (const float*)d_in[7];
    float* out = (float*)d_out;

    // workspace layout: msg[N*D] | h1[N*D] | deg[N] | colsum[128]
    float* msg    = (float*)d_ws;
    float* h1     = msg + ND;
    float* deg    = h1 + ND;
    float* colsum = deg + N_NODES;

    const int tileBlocks = N_NODES / 16;                // 6250 (exact)
    const int scatBlocks = (N_EDGES * 32 + 255) / 256;  // one wave32 per edge

    // zero accumulators (msg, deg, colsum) every call
    k_zero<<<2048, 256, 0, stream>>>((float4*)msg, (long)(ND / 4));
    k_zero<<<64,   256, 0, stream>>>((float4*)deg, (long)((N_NODES + D) / 4));

    // aggregate 1
    k_degree<<<2048, 256, 0, stream>>>(ei, deg);
    k_scatter<<<scatBlocks, 256, 0, stream>>>(nf, ei, msg);
    // layer 1: h1 = relu((nf + msg/deg) @ W1 + b1)
    k_layer1<<<tileBlocks, 256, 0, stream>>>(nf, msg, deg, W1, b1, h1);

    // aggregate 2 (reuse msg buffer)
    k_zero<<<2048, 256, 0, stream>>>((float4*)msg, (long)(ND / 4));
    k_scatter<<<scatBlocks, 256, 0, stream>>>(h1, ei, msg);
    // layer 2 fused with mean-readout accumulation
    k_layer2<<<tileBlocks, 256, 0, stream>>>(h1, msg, deg, W2, b2, colsum);

    // out = mean @ W3 + b3
    k_final<<<1, 128, 0, stream>>>(colsum, W3, b3, out);
}